// GraphSAGENodeModel_55997783605577
// MI455X (gfx1250) — compile-verified
//
#include <hip/hip_runtime.h>

#define IN_DIM  128
#define HID     256
#define OUT_DIM 64

typedef __attribute__((ext_vector_type(2))) float v2f;
typedef __attribute__((ext_vector_type(8))) float v8f;

// ---------------------------------------------------------------------------
// Edge scatter: agg[dst] += x[src], deg[dst] += 1  (mean numerator / denom)
// grid.x covers edges, grid.y covers feature quads (D/4). float4 gather,
// 4 global_atomic_add_f32 scatter (L2-resident at this problem size).
// ---------------------------------------------------------------------------
__global__ void edge_scatter_kernel(const float* __restrict__ x,
                                    const int* __restrict__ src,
                                    const int* __restrict__ dst,
                                    float* __restrict__ agg,
                                    float* __restrict__ deg,
                                    int D, int nE) {
    const int e = blockIdx.x * blockDim.x + threadIdx.x;
    if (e >= nE) return;
    const unsigned g4 = blockIdx.y * 4u;        // feature quad start
    const int s = src[e];
    const int d = dst[e];
    const float4 v = *reinterpret_cast<const float4*>(x + (unsigned)s * (unsigned)D + g4);
    float* p = agg + (unsigned)d * (unsigned)D + g4;
    atomicAdd(p + 0, v.x);
    atomicAdd(p + 1, v.y);
    atomicAdd(p + 2, v.z);
    atomicAdd(p + 3, v.w);
    if (deg != nullptr && g4 == 0u) atomicAdd(deg + d, 1.0f);
}

// deg[i] -> 1 / max(deg[i], 1)
__global__ void rdeg_kernel(float* __restrict__ deg, int n) {
    int i = blockIdx.x * blockDim.x + threadIdx.x;
    if (i < n) deg[i] = 1.0f / fmaxf(deg[i], 1.0f);
}

// ---------------------------------------------------------------------------
// Fused SAGE GEMM on native fp32 WMMA (V_WMMA_F32_16X16X4_F32):
//   Out = act( X @ Wl + (Agg * rdeg) @ Wr + bias )
// One wave computes a 32x64 output tile: 2 row-frags x 4 col-frags,
// 8 x v8f accumulators. Per k-step: 4 A loads + 8 B loads -> 8 WMMAs.
// All global accesses use 32-bit element offsets (base + zext(i32)) so the
// backend can use SGPR-base + 32-bit-VGPR-offset addressing; loads of a
// k-step are issued as one burst ahead of the WMMA chain.
//
// Fragment layouts (ISA 7.12.2, 32-bit):
//   A 16x4 : lane L(0..15) row M=L; v0/v1 = K = 2*half + {0,1}
//   B 4x16 : v0/v1 = rows K = 2*half + {0,1}, col N = lane&15
//   C 16x16: acc[v] = C[M = v + 8*half][N = lane&15]
// Agg==nullptr / do_relu are wave-uniform -> EXEC all-ones at every WMMA.
// ---------------------------------------------------------------------------
__global__ void sage_wmma_gemm_kernel(const float* __restrict__ X,
                                      const float* __restrict__ Agg,   // nullable
                                      const float* __restrict__ rdeg,  // nullable
                                      const float* __restrict__ Wl,
                                      const float* __restrict__ Wr,    // nullable
                                      const float* __restrict__ bias,
                                      float* __restrict__ Out,
                                      int Din, int Dout, int do_relu) {
    const int lane   = threadIdx.x & 31;
    const unsigned half   = (unsigned)(lane >> 4);
    const unsigned lane16 = (unsigned)(lane & 15);
    const unsigned row0 = blockIdx.x * 32u;
    const unsigned col0 = blockIdx.y * 64u;
    const unsigned uDin  = (unsigned)Din;
    const unsigned uDout = (unsigned)Dout;
    const unsigned rowStride16 = 16u * uDin;

    v8f acc[2][4];
#pragma unroll
    for (int r = 0; r < 2; ++r)
#pragma unroll
        for (int t = 0; t < 4; ++t) acc[r][t] = (v8f){};

    // ---- self term: X @ Wl ----
    {
        unsigned xo = (row0 + lane16) * uDin + 2u * half;        // A row offset
        unsigned wo = (2u * half) * uDout + col0 + lane16;       // B row offset
        for (int k = 0; k < Din; k += 4) {
            // load burst: A frags (contiguous pairs -> b64) then B frags
            v2f a0, a1, b[4];
            a0.x = X[xo];              a0.y = X[xo + 1u];
            a1.x = X[xo + rowStride16]; a1.y = X[xo + rowStride16 + 1u];
            const unsigned wo1 = wo + uDout;
#pragma unroll
            for (int t = 0; t < 4; ++t) {
                b[t].x = Wl[wo  + (unsigned)(t * 16)];
                b[t].y = Wl[wo1 + (unsigned)(t * 16)];
            }
#pragma unroll
            for (int t = 0; t < 4; ++t) {
                acc[0][t] = __builtin_amdgcn_wmma_f32_16x16x4_f32(
                    false, a0, false, b[t], (short)0, acc[0][t], false, false);
                acc[1][t] = __builtin_amdgcn_wmma_f32_16x16x4_f32(
                    false, a1, false, b[t], (short)0, acc[1][t], false, false);
            }
            xo += 4u;
            wo += 4u * uDout;
        }
    }

    // ---- neighbor term: (Agg * rdeg) @ Wr ----
    if (Agg != nullptr) {
        const float rd0 = rdeg[row0 + lane16];
        const float rd1 = rdeg[row0 + 16u + lane16];
        unsigned ao = (row0 + lane16) * uDin + 2u * half;
        unsigned wo = (2u * half) * uDout + col0 + lane16;
        for (int k = 0; k < Din; k += 4) {
            v2f a0, a1, b[4];
            a0.x = Agg[ao] * rd0;               a0.y = Agg[ao + 1u] * rd0;
            a1.x = Agg[ao + rowStride16] * rd1; a1.y = Agg[ao + rowStride16 + 1u] * rd1;
            const unsigned wo1 = wo + uDout;
#pragma unroll
            for (int t = 0; t < 4; ++t) {
                b[t].x = Wr[wo  + (unsigned)(t * 16)];
                b[t].y = Wr[wo1 + (unsigned)(t * 16)];
            }
#pragma unroll
            for (int t = 0; t < 4; ++t) {
                acc[0][t] = __builtin_amdgcn_wmma_f32_16x16x4_f32(
                    false, a0, false, b[t], (short)0, acc[0][t], false, false);
                acc[1][t] = __builtin_amdgcn_wmma_f32_16x16x4_f32(
                    false, a1, false, b[t], (short)0, acc[1][t], false, false);
            }
            ao += 4u;
            wo += 4u * uDout;
        }
    }

    // ---- bias + relu + store ----
#pragma unroll
    for (int r = 0; r < 2; ++r) {
        const unsigned obase = (row0 + (unsigned)r * 16u + 8u * half) * uDout
                             + col0 + lane16;
#pragma unroll
        for (int t = 0; t < 4; ++t) {
            const float bv = bias[col0 + (unsigned)(t * 16) + lane16];
#pragma unroll
            for (int v = 0; v < 8; ++v) {
                float val = acc[r][t][v] + bv;
                if (do_relu) val = fmaxf(val, 0.0f);
                Out[obase + (unsigned)v * uDout + (unsigned)(t * 16)] = val;
            }
        }
    }
}

// ---------------------------------------------------------------------------
// Host-side orchestration.
// inputs: 0:x 1:edge_index 2:Wl0 3:Wr0 4:b0 5:Wl1 6:Wr1 7:b1 8:Wf 9:bf
// d_out:  out[N*64] ++ node_embeddings[N*256]
// ---------------------------------------------------------------------------
extern "C" void kernel_launch(void* const* d_in, const int* in_sizes, int n_in,
                              void* d_out, int out_size, void* d_ws, size_t ws_size,
                              hipStream_t stream) {
    const float* x   = (const float*)d_in[0];
    const int*   ei  = (const int*)d_in[1];
    const float* Wl0 = (const float*)d_in[2];
    const float* Wr0 = (const float*)d_in[3];
    const float* b0  = (const float*)d_in[4];
    const float* Wl1 = (const float*)d_in[5];
    const float* Wr1 = (const float*)d_in[6];
    const float* b1  = (const float*)d_in[7];
    const float* Wf  = (const float*)d_in[8];
    const float* bf  = (const float*)d_in[9];

    const int N = in_sizes[0] / IN_DIM;        // 100000 (divisible by 32)
    const int E = in_sizes[1] / 2;             // 3200000

    const int* src = ei;          // edge_index[0]
    const int* dst = ei + E;      // edge_index[1]

    float* out = (float*)d_out;                       // [N, OUT_DIM]
    float* emb = out + (size_t)N * OUT_DIM;           // [N, HID] node embeddings

    // workspace layout: deg | agg (N*HID) | h0 (N*HID)
    char*  ws  = (char*)d_ws;
    size_t off = 0;
    float* deg = (float*)(ws + off);
    off += ((size_t)N * sizeof(float) + 255) & ~(size_t)255;
    float* agg = (float*)(ws + off);
    off += ((size_t)N * HID * sizeof(float) + 255) & ~(size_t)255;
    float* h0  = (float*)(ws + off);

    const int eb  = 256;
    const int ebx = (E + eb - 1) / eb;

    // ===== Layer 0 =====
    hipMemsetAsync(deg, 0, (size_t)N * sizeof(float), stream);
    hipMemsetAsync(agg, 0, (size_t)N * IN_DIM * sizeof(float), stream);
    edge_scatter_kernel<<<dim3(ebx, IN_DIM / 4), eb, 0, stream>>>(
        x, src, dst, agg, deg, IN_DIM, E);
    rdeg_kernel<<<(N + 255) / 256, 256, 0, stream>>>(deg, N);
    sage_wmma_gemm_kernel<<<dim3(N / 32, HID / 64), 32, 0, stream>>>(
        x, agg, deg, Wl0, Wr0, b0, h0, IN_DIM, HID, 1);

    // ===== Layer 1 =====
    hipMemsetAsync(agg, 0, (size_t)N * HID * sizeof(float), stream);
    edge_scatter_kernel<<<dim3(ebx, HID / 4), eb, 0, stream>>>(
        h0, src, dst, agg, nullptr, HID, E);
    sage_wmma_gemm_kernel<<<dim3(N / 32, HID / 64), 32, 0, stream>>>(
        h0, agg, deg, Wl1, Wr1, b1, emb, HID, HID, 1);

    // ===== Final linear: out = emb @ Wf + bf =====
    sage_wmma_gemm_kernel<<<dim3(N / 32, OUT_DIM / 64), 32, 0, stream>>>(
        emb, nullptr, nullptr, Wf, nullptr, bf, out, HID, OUT_DIM, 0);
}